// DiagonalLinear_37941741093042
// MI455X (gfx1250) — compile-verified
//
#include <hip/hip_runtime.h>
#include <cmath>

// DiagonalLinear: out[t, c] = x[t, c] * d[c],  d = threshold(diag(W))
// Pure streaming problem: 256 MiB of traffic @ 23.3 TB/s -> ~11 us floor.
// WMMA is useless at 0.125 FLOP/byte; the CDNA5 features that matter are the
// async-LDS data path (ASYNCcnt) and NT cache hints for the streamed tensors.

#define NCOL    4096
#define THRESH  1e-4f

typedef __attribute__((ext_vector_type(4))) float v4f;
typedef __attribute__((ext_vector_type(4))) int   v4i;

// Address-space-qualified vector types for the async builtin:
// clang declares __builtin_amdgcn_global_load_async_to_lds_b128 as
// (AS1 int4* src, AS3 int4* dst, int offset, int cpol).
typedef __attribute__((address_space(1))) v4i gv4i;  // global
typedef __attribute__((address_space(3))) v4i lv4i;  // LDS

#if defined(__has_builtin)
#  if __has_builtin(__builtin_amdgcn_global_load_async_to_lds_b128)
#    define HAVE_ASYNC_B128 1
#  endif
#  if __has_builtin(__builtin_amdgcn_s_wait_asynccnt)
#    define HAVE_WAIT_ASYNC 1
#  endif
#endif

__device__ __forceinline__ void wait_async_zero() {
#if defined(__AMDGCN__)
#  if defined(HAVE_WAIT_ASYNC)
    __builtin_amdgcn_s_wait_asynccnt(0);
#  else
    asm volatile("s_wait_asynccnt 0" ::: "memory");
#  endif
#endif
}

// Kernel 1: gather + threshold the diagonal of W into workspace.
// d[i] = W[i*(N+1)] if |W[i*(N+1)]| > THRESH else 0.
__global__ void diag_extract_kernel(const float* __restrict__ W,
                                    float* __restrict__ dws, int n) {
    int i = blockIdx.x * blockDim.x + threadIdx.x;
    if (i < n) {
        float v = W[(size_t)i * (size_t)(n + 1)];
        dws[i] = (fabsf(v) > THRESH) ? v : 0.0f;
    }
}

// Kernel 2: stream rows. 1024 threads cover one full 4096-col row (float4 per
// thread). Diagonal staged into LDS via the CDNA5 async-to-LDS path, then held
// in registers for the whole grid-stride row loop.
__global__ void __launch_bounds__(1024)
diag_scale_kernel(const float* __restrict__ x,
                  const float* __restrict__ dws,
                  float* __restrict__ out, int tokens) {
    __shared__ float sd[NCOL];
    const int t = threadIdx.x;  // 0..1023, owns columns [4t, 4t+4)

#if defined(__AMDGCN__) && defined(HAVE_ASYNC_B128)
    // One async b128 per lane: 1024 lanes x 16B = 16 KB diagonal -> LDS.
    __builtin_amdgcn_global_load_async_to_lds_b128(
        (gv4i*)(dws + 4 * t), (lv4i*)(sd + 4 * t), 0, 0);
    wait_async_zero();
#else
    ((v4f*)sd)[t] = ((const v4f*)dws)[t];
#endif
    __syncthreads();

    const v4f dv = ((const v4f*)sd)[t];  // ds_load_b128; lives in VGPRs after

    const size_t colb = 4 * (size_t)t;
    for (int row = blockIdx.x; row < tokens; row += gridDim.x) {
        size_t off = (size_t)row * NCOL + colb;
        v4f xv = __builtin_nontemporal_load((const v4f*)(x + off));   // NT: read-once stream
        v4f ov = xv * dv;                                             // 4x v_mul_f32
        __builtin_nontemporal_store(ov, (v4f*)(out + off));           // NT: write-once stream
    }
}

extern "C" void kernel_launch(void* const* d_in, const int* in_sizes, int n_in,
                              void* d_out, int out_size, void* d_ws, size_t ws_size,
                              hipStream_t stream) {
    const float* x = (const float*)d_in[0];   // [TOKENS, N] fp32
    const float* W = (const float*)d_in[1];   // [N, N] fp32
    float* out     = (float*)d_out;           // [TOKENS, N] fp32
    float* dws     = (float*)d_ws;            // N floats of scratch (16 KB)

    const int n      = NCOL;
    const int tokens = in_sizes[0] / n;       // 8192

    diag_extract_kernel<<<(n + 255) / 256, 256, 0, stream>>>(W, dws, n);

    int grid = tokens < 1024 ? tokens : 1024; // 1024 blocks x 32 waves: fills the chip
    diag_scale_kernel<<<grid, 1024, 0, stream>>>(x, dws, out, tokens);
}